// TonalDiffusionModel_79731772883080
// MI455X (gfx1250) — compile-verified
//
#include <hip/hip_runtime.h>
#include <math.h>

// ---------------------------------------------------------------------------
// Tonal diffusion model, MI455X (gfx1250, wave32).
// One wave (32 lanes) processes one data row end-to-end.
//   Phase A: compound-Poisson stencil diffusion over S=256 support.
//            LDS ping-pong with 4-elem zero halos (no boundary guards) and
//            uniform support-window skipping (run_n support = [128-4n,128+4n]).
//   Phase B: KL matching over 129 shifts. cross[k] = sum_c data[c]*log(acc[k+c])
//            via V_WMMA_F32_16X16X4_F32 with an im2col/Toeplitz mapping:
//            A[m,j]=data[j-m], B[j,n]=logacc[j+16n]  =>  D[m,n]=cross[m+16n].
// ---------------------------------------------------------------------------

typedef __attribute__((ext_vector_type(2))) float v2f;
typedef __attribute__((ext_vector_type(8))) float v8f;

#ifndef __has_builtin
#define __has_builtin(x) 0
#endif

#if __has_builtin(__builtin_amdgcn_wmma_f32_16x16x4_f32)
#define USE_WMMA_F32X4 1
#else
#define USE_WMMA_F32X4 0
#endif

namespace {
constexpr int kS = 256;        // distribution support
constexpr int kC = 128;        // data columns
constexpr int kCenter = 128;   // delta start position
constexpr int kIters = 33;     // Poisson scan iterations
constexpr int kShifts = 129;   // S - C + 1
constexpr int kKTiles = 36;    // ceil((128+15)/4) K-tiles for the WMMA correlation
constexpr int kRows = 8;       // rows (waves) per block
constexpr int kHalo = 4;       // max |interval step|
constexpr int kRunBuf = kS + 2 * kHalo;  // 264
}

__global__ __launch_bounds__(256) void tonal_diffusion_kernel(
    const float* __restrict__ data,        // [n,128]
    const float* __restrict__ log_weights, // [n,6]
    float* __restrict__ out,               // [n]
    int n_rows) {
  const int lane = threadIdx.x & 31;
  const int wave = threadIdx.x >> 5;
  const int row  = blockIdx.x * kRows + wave;
  const int rowc = (row < n_rows) ? row : (n_rows - 1);  // clamp, keep EXEC uniform

  __shared__ float sRunA[kRows][kRunBuf];  // zero halos at [0,4) and [260,264)
  __shared__ float sRunB[kRows][kRunBuf];
  __shared__ float sDataPad[kRows][160];   // 16 zero | 128 data | 16 zero
  __shared__ float sLogAcc[kRows][384];    // log(acc) padded with zeros above 255
  __shared__ float sZ[kRows][132];         // log of sliding-window sums (129 used)

  const int steps[6] = {1, -1, -3, 3, 4, -4};

  // --- per-row interval probabilities & Poisson rate (scalar, all lanes) ----
  float p[6];
  float rate = 0.f;
#pragma unroll
  for (int i = 0; i < 6; ++i) {
    p[i] = expf(log_weights[(size_t)rowc * 6 + i]);
    rate += p[i];
  }
  const float inv_rate = 1.f / rate;
#pragma unroll
  for (int i = 0; i < 6; ++i) p[i] *= inv_rate;

  // --- stage padded data row (tiny entries treated as exact zeros, per isclose)
  for (int idx = lane; idx < 160; idx += 32) {
    float v = 0.f;
    if (idx >= 16 && idx < 16 + kC) {
      float raw = data[(size_t)rowc * kC + (idx - 16)];
      v = (raw > 1e-8f) ? raw : 0.f;
    }
    sDataPad[wave][idx] = v;
  }

  // --- Phase A: diffusion ---------------------------------------------------
  // Zero-init BOTH buffers (incl. halos); with window-skipped writes, anything
  // outside the current support stays exactly zero in both buffers.
  for (int idx = lane; idx < kRunBuf; idx += 32) {
    sRunA[wave][idx] = 0.f;
    sRunB[wave][idx] = 0.f;
  }
  float* runA = &sRunA[wave][kHalo];  // index by t in [-4, 260)
  float* runB = &sRunB[wave][kHalo];
  if (lane == 0) runA[kCenter] = 1.f;

  float acc[8];
#pragma unroll
  for (int e = 0; e < 8; ++e) acc[e] = 0.f;
  __syncthreads();

  float sp = expf(-rate);  // Poisson pmf(0); exact recurrence below
  for (int n = 0; n < kIters; ++n) {
    // support of run_{n+1}: [128-4(n+1), 128+4(n+1)] clipped to [0,255]
    const int lo_t = kCenter - kHalo * (n + 1);
    const int hi_t = kCenter + kHalo * (n + 1);
    const int eLo = (lo_t > 0 ? lo_t : 0) >> 5;
    const int eHi = (hi_t < kS - 1 ? hi_t : kS - 1) >> 5;
#pragma unroll
    for (int e = 0; e < 8; ++e) {
      if (e >= eLo && e <= eHi) {  // uniform (scalar) predicate
        const int t = e * 32 + lane;
        const float r = runA[t];
        acc[e] = fmaf(sp, r, acc[e]);
        float nv = 0.f;
#pragma unroll
        for (int i = 0; i < 6; ++i)
          nv = fmaf(p[i], runA[t - steps[i]], nv);  // halo reads give 0
        runB[t] = nv;
      }
    }
    __syncthreads();
    float* tmp = runA; runA = runB; runB = tmp;
    sp = sp * rate / (float)(n + 1);
  }

  // --- stage acc + log(acc) in LDS -----------------------------------------
  float* accL = runA;  // run buffers are dead now; reuse one for acc
#pragma unroll
  for (int e = 0; e < 8; ++e) {
    const int t = e * 32 + lane;
    accL[t] = acc[e];
    sLogAcc[wave][t] = logf(fmaxf(acc[e], 1e-38f));
  }
  for (int t = kS + lane; t < 384; t += 32) sLogAcc[wave][t] = 0.f;
  __syncthreads();

  // --- sliding-window normalizers: Z[k] = log(sum_c acc[k+c]) --------------
  for (int k = lane; k < kShifts; k += 32) {
    float z = 0.f;
    for (int c = 0; c < kC; ++c) z += accL[k + c];
    sZ[wave][k] = logf(z);
  }

  // --- plogp = sum_c data*log(data) (wave reduction) ------------------------
  float pl = 0.f;
  for (int c = lane; c < kC; c += 32) {
    const float v = sDataPad[wave][16 + c];
    if (v > 0.f) pl += v * logf(v);
  }
#pragma unroll
  for (int off = 16; off; off >>= 1) pl += __shfl_xor(pl, off, 32);
  __syncthreads();  // Z visible to all lanes

  float best = INFINITY;

#if USE_WMMA_F32X4
  // --- correlation via fp32 WMMA 16x16x4 ------------------------------------
  // A fragment (16x4 fp32): lane m=lane&15; VGPR v holds K = (lane<16?0:2)+v.
  //   A[m,j] = dataPad16[j-m]           -> dp[j+v], dp = &sDataPad[16-m]
  // B fragment (4x16 fp32): lane n=lane&15; VGPR v holds K = (lane<16?0:2)+v.
  //   B[j,n] = logaccPad[j+16n]         -> lp[j+v], lp = &sLogAcc[16n]
  // D[m,n] = sum_j A[m,j]B[j,n] = cross[m+16n].
  const int m    = lane & 15;
  const int koff = (lane & 16) ? 2 : 0;
  const float* dp = &sDataPad[wave][16 - m];
  const float* lp = &sLogAcc[wave][16 * m];
  v8f dacc = {0.f, 0.f, 0.f, 0.f, 0.f, 0.f, 0.f, 0.f};
#pragma unroll
  for (int kc = 0; kc < kKTiles; ++kc) {
    const int j = 4 * kc + koff;
    v2f a = {dp[j], dp[j + 1]};
    v2f b = {lp[j], lp[j + 1]};
    dacc = __builtin_amdgcn_wmma_f32_16x16x4_f32(
        /*neg_a=*/false, a, /*neg_b=*/false, b,
        /*c_mod=*/(short)0, dacc, /*reuse_a=*/false, /*reuse_b=*/false);
  }
  // D layout: VGPR r -> M = r (lanes 0-15) or r+8 (lanes 16-31); N = lane&15.
  const int mbase = (lane & 16) ? 8 : 0;
  const int n16   = 16 * (lane & 15);
#pragma unroll
  for (int r = 0; r < 8; ++r) {
    const int k = (r + mbase) + n16;
    if (k < kShifts) {
      const float loss = pl - dacc[r] + sZ[wave][k];
      best = fminf(best, loss);
    }
  }
#else
  // --- VALU fallback correlation -------------------------------------------
  for (int k = lane; k < kShifts; k += 32) {
    float cr = 0.f;
    for (int c = 0; c < kC; ++c)
      cr = fmaf(sDataPad[wave][16 + c], sLogAcc[wave][k + c], cr);
    best = fminf(best, pl - cr + sZ[wave][k]);
  }
#endif

#pragma unroll
  for (int off = 16; off; off >>= 1) best = fminf(best, __shfl_xor(best, off, 32));
  if (lane == 0 && row < n_rows) out[row] = best;
}

extern "C" void kernel_launch(void* const* d_in, const int* in_sizes, int n_in,
                              void* d_out, int out_size, void* d_ws, size_t ws_size,
                              hipStream_t stream) {
  (void)n_in; (void)d_ws; (void)ws_size; (void)out_size;
  const float* data = (const float*)d_in[0];          // [n,128] f32
  const float* logw = (const float*)d_in[1];          // [n,6]   f32
  float* out = (float*)d_out;                         // [n]     f32
  const int n_rows = in_sizes[0] / kC;
  const int blocks = (n_rows + kRows - 1) / kRows;
  tonal_diffusion_kernel<<<blocks, 256, 0, stream>>>(data, logw, out, n_rows);
}